// FraudDetectionNet_65481071396090
// MI455X (gfx1250) — compile-verified
//
#include <hip/hip_runtime.h>
#include <hip/hip_bf16.h>

typedef __attribute__((ext_vector_type(2))) float v2f;
typedef __attribute__((ext_vector_type(8))) float v8f;

#define WIRES 4
#define DIM 16
#define NPATCH 196

// Simulate one 4-qubit patch circuit entirely in registers and return the
// partial dot product  sum_q z_q * w_q.
__device__ __forceinline__ float patch_feature_dot(float p00, float p01,
                                                   float p10, float p11,
                                                   float4 wv) {
    const float HALF_PI = 1.5707963267948966f;
    float ang[4] = {p00, p01, p10, p11};
    float c[4], s[4];
#pragma unroll
    for (int q = 0; q < 4; ++q) {
        // half-angle = clip(p,-1,1)*pi*0.5
        float h = fminf(fmaxf(ang[q], -1.0f), 1.0f) * HALF_PI;
        s[q] = __sinf(h);
        c[q] = __cosf(h);
    }

    float st[DIM];
#pragma unroll
    for (int b = 0; b < DIM; ++b) st[b] = 0.0f;
    st[0] = 1.0f;

    // Sequence: RY layer; then 3x (RY layer, CNOT(0,1), CNOT(2,3)).
#pragma unroll
    for (int layer = 0; layer < 4; ++layer) {
#pragma unroll
        for (int q = 0; q < 4; ++q) {
            const int step = 1 << (3 - q);   // wire q acts on bit (3-q)
#pragma unroll
            for (int b = 0; b < DIM; ++b) {
                if ((b & step) == 0) {
                    float a0 = st[b], a1 = st[b + step];
                    st[b]        = c[q] * a0 - s[q] * a1;
                    st[b + step] = s[q] * a0 + c[q] * a1;
                }
            }
        }
        if (layer != 0) {
            // CNOT(c=0,t=1): control bit3, flip bit2 -> swap (b, b^4), b=8..11
#pragma unroll
            for (int b = 8; b < 12; ++b) {
                float tmp = st[b]; st[b] = st[b + 4]; st[b + 4] = tmp;
            }
            // CNOT(c=2,t=3): control bit1, flip bit0 -> swap (b, b^1) for bit1=1
#pragma unroll
            for (int b = 0; b < DIM; ++b) {
                if ((b & 2) && !(b & 1)) {
                    float tmp = st[b]; st[b] = st[b + 1]; st[b + 1] = tmp;
                }
            }
        }
    }

    float pr[DIM];
#pragma unroll
    for (int b = 0; b < DIM; ++b) pr[b] = st[b] * st[b];

    float dot = 0.0f;
#pragma unroll
    for (int q = 0; q < 4; ++q) {
        const int bit = 1 << (3 - q);
        float z = 0.0f;
#pragma unroll
        for (int b = 0; b < DIM; ++b) z += (b & bit) ? -pr[b] : pr[b];
        float w = (q == 0) ? wv.x : (q == 1) ? wv.y : (q == 2) ? wv.z : wv.w;
        dot += z * w;
    }
    return dot;
}

// Kernel 1: one thread per (image, patch). Writes partial[img*196 + t].
__global__ void __launch_bounds__(224)
patch_kernel(const float* __restrict__ x, const float* __restrict__ W,
             float* __restrict__ partial) {
    const int img = blockIdx.x;
    const int t = threadIdx.x;
    if (t >= NPATCH) return;
    const int i = t / 14, j = t % 14;
    const float* px = x + (size_t)img * 784 + i * 56 + j * 2;
    float p00 = px[0], p01 = px[1], p10 = px[28], p11 = px[29];
    const float4 wv = *reinterpret_cast<const float4*>(W + 4 * t);
    partial[(size_t)img * NPATCH + t] = patch_feature_dot(p00, p01, p10, p11, wv);
}

// Kernel 2: one wave32 per 16 images. Reduces 196 partials per image via
// chained V_WMMA_F32_16X16X4_F32 (A = 16x4 partial chunk, B = ones), then
// sigmoid. Full f32 precision throughout.
__global__ void __launch_bounds__(32)
reduce_wmma_kernel(const float* __restrict__ partial,
                   const float* __restrict__ bias,
                   float* __restrict__ out) {
    const int lane = threadIdx.x;            // 0..31
    const int imgBase = blockIdx.x * 16;
    const int m = lane & 15;                 // A-matrix row (image within group)
    const int koff = (lane >> 4) << 1;       // lanes 0-15 -> K=0,1 ; 16-31 -> K=2,3
    const float* row = partial + (size_t)(imgBase + m) * NPATCH + koff;

    v8f acc = {};
    v2f bones; bones[0] = 1.0f; bones[1] = 1.0f;   // B = 4x16 ones (layout-invariant)

    for (int chunk = 0; chunk < NPATCH / 4; ++chunk) {   // 49 chunks of K=4
        v2f a = *reinterpret_cast<const v2f*>(row + 4 * chunk); // b64: A[m][koff..koff+1]
        acc = __builtin_amdgcn_wmma_f32_16x16x4_f32(
            /*neg_a=*/false, a, /*neg_b=*/false, bones,
            /*c_mod=*/(short)0, acc, /*reuse_a=*/false, /*reuse_b=*/false);
    }

    // D[m][n] = sum_p partial[m][p], identical across n. Lane 0 (N=0) holds
    // rows 0..7 in acc[0..7]; lane 16 holds rows 8..15.
    const float bval = bias[0];
    if (m == 0) {
        const int rowBase = imgBase + ((lane >> 4) << 3);
#pragma unroll
        for (int r = 0; r < 8; ++r) {
            float v = acc[r] + bval;
            out[rowBase + r] = 1.0f / (1.0f + __expf(-v));
        }
    }
}

// Fallback (if workspace is too small or B % 16 != 0): fused per-image LDS reduction.
__global__ void __launch_bounds__(256)
fused_kernel(const float* __restrict__ x, const float* __restrict__ W,
             const float* __restrict__ bias, float* __restrict__ out) {
    __shared__ float red[256];
    const int img = blockIdx.x;
    const int t = threadIdx.x;
    float val = 0.0f;
    if (t < NPATCH) {
        const int i = t / 14, j = t % 14;
        const float* px = x + (size_t)img * 784 + i * 56 + j * 2;
        const float4 wv = *reinterpret_cast<const float4*>(W + 4 * t);
        val = patch_feature_dot(px[0], px[1], px[28], px[29], wv);
    }
    red[t] = val;
    __syncthreads();
#pragma unroll
    for (int off = 128; off > 0; off >>= 1) {
        if (t < off) red[t] += red[t + off];
        __syncthreads();
    }
    if (t == 0) {
        float v = red[0] + bias[0];
        out[img] = 1.0f / (1.0f + __expf(-v));
    }
}

extern "C" void kernel_launch(void* const* d_in, const int* in_sizes, int n_in,
                              void* d_out, int out_size, void* d_ws, size_t ws_size,
                              hipStream_t stream) {
    const float* x = (const float*)d_in[0];   // (B,1,28,28)
    const float* W = (const float*)d_in[1];   // (784,1)
    const float* b = (const float*)d_in[2];   // (1,)
    float* out = (float*)d_out;               // (B,1) float32

    const int B = in_sizes[0] / 784;
    const size_t need = (size_t)B * NPATCH * sizeof(float);

    if (ws_size >= need && (B % 16) == 0) {
        float* partial = (float*)d_ws;
        patch_kernel<<<B, 224, 0, stream>>>(x, W, partial);
        reduce_wmma_kernel<<<B / 16, 32, 0, stream>>>(partial, b, out);
    } else {
        fused_kernel<<<B, 256, 0, stream>>>(x, W, b, out);
    }
}